// GCNEncoder_68204080660517
// MI455X (gfx1250) — compile-verified
//
#include <hip/hip_runtime.h>
#include <hip/hip_bf16.h>

typedef __attribute__((ext_vector_type(2))) float v2f;
typedef __attribute__((ext_vector_type(8))) float v8f;

// ---------------------------------------------------------------------------
// Zero-fill (float4 vectorized). Needed before atomic scatter-accumulate.
// ---------------------------------------------------------------------------
__global__ void gcn_zero_kernel(float4* __restrict__ p, long n4) {
    long i = (long)blockIdx.x * blockDim.x + threadIdx.x;
    if (i < n4) p[i] = make_float4(0.f, 0.f, 0.f, 0.f);
}

// ---------------------------------------------------------------------------
// Edge-parallel SpMM: out[rows[e]] += vals[e] * x[cols[e]]   (F floats/row)
// One wave32 per edge. F=256 -> 8 floats/lane (2x b128 load, 8 f32 atomics).
// F=128 -> 4 floats/lane (1x b128 load, 4 f32 atomics).
// x (102MB) / h1 (51MB) are L2-resident on MI455X (192MB L2), so both the
// gather and the atomic adds are serviced by L2.
// ---------------------------------------------------------------------------
template<int F>
__global__ void gcn_spmm_kernel(const int* __restrict__ rows,
                                const int* __restrict__ cols,
                                const float* __restrict__ vals,
                                const float* __restrict__ x,
                                float* __restrict__ out,
                                int n_edges) {
    int gid  = blockIdx.x * blockDim.x + threadIdx.x;
    int e    = gid >> 5;
    int lane = gid & 31;
    if (e >= n_edges) return;

    int   r = rows[e];
    int   c = cols[e];
    float v = vals[e];

    const float* src = x   + (size_t)c * F;
    float*       dst = out + (size_t)r * F;

    if (F == 256) {
        int o = lane * 8;
        float4 p = *(const float4*)(src + o);
        float4 q = *(const float4*)(src + o + 4);
        atomicAdd(dst + o + 0, v * p.x);
        atomicAdd(dst + o + 1, v * p.y);
        atomicAdd(dst + o + 2, v * p.z);
        atomicAdd(dst + o + 3, v * p.w);
        atomicAdd(dst + o + 4, v * q.x);
        atomicAdd(dst + o + 5, v * q.y);
        atomicAdd(dst + o + 6, v * q.z);
        atomicAdd(dst + o + 7, v * q.w);
    } else { // F == 128
        int o = lane * 4;
        float4 p = *(const float4*)(src + o);
        atomicAdd(dst + o + 0, v * p.x);
        atomicAdd(dst + o + 1, v * p.y);
        atomicAdd(dst + o + 2, v * p.z);
        atomicAdd(dst + o + 3, v * p.w);
    }
}

// ---------------------------------------------------------------------------
// Dense layer via V_WMMA_F32_16X16X4_F32 (exact f32 semantics, wave32).
// out[M,N] = act(A[M,K] @ B[K,N] + bias[N])
// Grid: one block per 16-row M-tile; one wave per 16-col N-tile (N/16 waves).
// All dims are exact multiples (M=100000=6250*16), so EXEC is all-1s at every
// WMMA as the ISA requires.
//
// A-frag (16x4 f32, ISA 7.12.2): lane<16 -> row m0+lane, K={k,k+1};
//                                lane>=16 -> K={k+2,k+3}  => one b64 load.
// B-frag (4x16):  lane l -> col n0+(l&15), rows k+2*(l>>4), +1.
// C/D:            VGPR r -> M = m0 + 8*(l>>4) + r, N = n0 + (l&15).
// ---------------------------------------------------------------------------
template<int K, int N, bool RELU>
__global__ void gcn_gemm_wmma_kernel(const float* __restrict__ A,
                                     const float* __restrict__ B,
                                     const float* __restrict__ bias,
                                     float* __restrict__ out) {
    const int wave = threadIdx.x >> 5;
    const int lane = threadIdx.x & 31;
    const int m0   = blockIdx.x * 16;
    const int n0   = wave * 16;
    const int half = lane >> 4;   // 0: lanes 0-15, 1: lanes 16-31
    const int l15  = lane & 15;

    const float* arow = A + (size_t)(m0 + l15) * K;
    const float* bcol = B + n0 + l15;

    v8f acc = {};
#pragma unroll 8
    for (int k = 0; k < K; k += 4) {
        const int ka = k + half * 2;            // even -> 8B aligned b64 load
        v2f a = *(const v2f*)(arow + ka);
        v2f b;
        b.x = bcol[(size_t)ka * N];
        b.y = bcol[(size_t)(ka + 1) * N];
        // 8 args: (neg_a, A, neg_b, B, c_mod, C, reuse_a, reuse_b)
        acc = __builtin_amdgcn_wmma_f32_16x16x4_f32(
            false, a, false, b, (short)0, acc, false, false);
    }

    const float bv    = bias[n0 + l15];
    const int   mbase = m0 + half * 8;
#pragma unroll
    for (int r = 0; r < 8; ++r) {
        float v = acc[r] + bv;
        if (RELU) v = fmaxf(v, 0.f);
        out[(size_t)(mbase + r) * N + (n0 + l15)] = v;
    }
}

// ---------------------------------------------------------------------------
// kernel_launch: zero(h0) -> spmm256 -> wmma 256->128 +ReLU
//             -> zero(h2) -> spmm128 -> wmma 128->64
// ---------------------------------------------------------------------------
extern "C" void kernel_launch(void* const* d_in, const int* in_sizes, int n_in,
                              void* d_out, int out_size, void* d_ws, size_t ws_size,
                              hipStream_t stream) {
    const float* x    = (const float*)d_in[0]; // [N_NODES, 256]
    const int*   rows = (const int*)  d_in[1]; // [N_EDGES]
    const int*   cols = (const int*)  d_in[2]; // [N_EDGES]
    const float* vals = (const float*)d_in[3]; // [N_EDGES]
    const float* W1   = (const float*)d_in[4]; // [256,128]
    const float* b1   = (const float*)d_in[5]; // [128]
    const float* W2   = (const float*)d_in[6]; // [128,64]
    const float* b2   = (const float*)d_in[7]; // [64]
    float*       out  = (float*)d_out;         // [N_NODES, 64]

    const int n_nodes = in_sizes[0] / 256;     // 100000
    const int n_edges = in_sizes[1];           // 3200000

    float* h0 = (float*)d_ws;                       // [n_nodes,256] spmm1 acc
    float* h1 = h0 + (size_t)n_nodes * 256;         // [n_nodes,128] layer1 out
    float* h2 = h1 + (size_t)n_nodes * 128;         // [n_nodes,128] spmm2 acc

    const int  mtiles     = n_nodes / 16;           // 6250 (exact)
    const int  spmm_grid  = (n_edges + 7) / 8;      // 8 waves (edges) / block
    const long h0_f4      = (long)n_nodes * 256 / 4;
    const long h2_f4      = (long)n_nodes * 128 / 4;

    // Layer 0: h0 = spmm(x)
    gcn_zero_kernel<<<(int)((h0_f4 + 255) / 256), 256, 0, stream>>>((float4*)h0, h0_f4);
    gcn_spmm_kernel<256><<<spmm_grid, 256, 0, stream>>>(rows, cols, vals, x, h0, n_edges);
    // h1 = relu(h0 @ W1 + b1)   [K=256, N=128] -> 8 waves/block
    gcn_gemm_wmma_kernel<256, 128, true><<<mtiles, 256, 0, stream>>>(h0, W1, b1, h1);

    // Layer 1: h2 = spmm(h1)
    gcn_zero_kernel<<<(int)((h2_f4 + 255) / 256), 256, 0, stream>>>((float4*)h2, h2_f4);
    gcn_spmm_kernel<128><<<spmm_grid, 256, 0, stream>>>(rows, cols, vals, h1, h2, n_edges);
    // out = h2 @ W2 + b2        [K=128, N=64] -> 4 waves/block
    gcn_gemm_wmma_kernel<128, 64, false><<<mtiles, 128, 0, stream>>>(h2, W2, b2, out);
}